// NTMHead_23630910062995
// MI455X (gfx1250) — compile-verified
//
#include <hip/hip_runtime.h>
#include <math.h>

#define BB 4096
#define NN 8192
#define WW 64

typedef float v2f __attribute__((ext_vector_type(2)));
typedef float v8f __attribute__((ext_vector_type(8)));

// ---------------------------------------------------------------------------
// Kernel 1: row L2-normalizations.
//   blocks [0, BB)        : A_scaled[b][w] = beta[b] * k_c[b][w]
//   blocks [BB, BB+NN)    : Mc[n][w]       = memory[n][w] / max(||memory[n]||,1e-8)
// One 64-thread block per row; LDS tree reduction (deterministic).
// ---------------------------------------------------------------------------
__global__ __launch_bounds__(64) void ntm_prep_kernel(
    const float* __restrict__ k, const float* __restrict__ beta,
    const float* __restrict__ memory,
    float* __restrict__ Asc, float* __restrict__ Mc)
{
  __shared__ float red[64];
  const int row = blockIdx.x;
  const int t = threadIdx.x;
  if (row < BB) {
    float x = k[row * WW + t];
    red[t] = x * x;
    for (int s2 = 32; s2 > 0; s2 >>= 1) { __syncthreads(); if (t < s2) red[t] += red[t + s2]; }
    __syncthreads();
    const float n1 = sqrtf(red[0]);
    __syncthreads();
    const float kn = x / fmaxf(n1, 1e-12f);
    red[t] = kn * kn;
    for (int s2 = 32; s2 > 0; s2 >>= 1) { __syncthreads(); if (t < s2) red[t] += red[t + s2]; }
    __syncthreads();
    const float n2 = sqrtf(red[0]);
    const float kc = kn / fmaxf(n2, 1e-8f);
    Asc[row * WW + t] = beta[row] * kc;   // fold beta into A
  } else {
    const int r = row - BB;
    float x = memory[r * WW + t];
    red[t] = x * x;
    for (int s2 = 32; s2 > 0; s2 >>= 1) { __syncthreads(); if (t < s2) red[t] += red[t + s2]; }
    __syncthreads();
    const float nm = sqrtf(red[0]);
    Mc[r * WW + t] = x / fmaxf(nm, 1e-8f);
  }
}

// ---------------------------------------------------------------------------
// Kernel 2: E[b][n] = exp( dot(Asc[b], Mc[n]) )  via V_WMMA_F32_16X16X4_F32.
// One wave per 16x16 C tile; 16 WMMAs cover K=64.
// A frag (16x4 f32): lane = M (lane&15), VGPR j holds K = 2*(lane>>4)+j.
// B frag (4x16 f32): lane = N (lane&15), VGPR j holds K = 2*(lane>>4)+j,
//   where B[k][n] = Mc[n][k].
// C/D (16x16 f32, 8 VGPRs): lanes 0-15 -> M=v, N=lane; lanes 16-31 -> M=v+8.
// ---------------------------------------------------------------------------
__global__ __launch_bounds__(256) void ntm_gemm_exp_kernel(
    const float* __restrict__ Asc, const float* __restrict__ Mc,
    float* __restrict__ E)
{
  const int lane  = threadIdx.x & 31;
  const int wave  = threadIdx.x >> 5;
  const int ntile = (blockIdx.x * 8 + wave) << 4;   // N tile base
  const int mtile = blockIdx.y << 4;                // M tile base
  const int l15   = lane & 15;
  const int khalf = (lane >> 4) << 1;               // 0 or 2

  const float* __restrict__ arow = Asc + (size_t)(mtile + l15) * WW + khalf;
  const float* __restrict__ brow = Mc  + (size_t)(ntile + l15) * WW + khalf;

  v8f c = {};
#pragma unroll
  for (int k0 = 0; k0 < WW; k0 += 4) {
    v2f a, b;
    a.x = arow[k0];     a.y = arow[k0 + 1];
    b.x = brow[k0];     b.y = brow[k0 + 1];
    c = __builtin_amdgcn_wmma_f32_16x16x4_f32(
        /*neg_a=*/false, a, /*neg_b=*/false, b,
        /*c_mod=*/(short)0, c, /*reuse_a=*/false, /*reuse_b=*/false);
  }

  // logits are in [-1,1] (beta in [0,1), cosine in [-1,1]) -> exp is safe
  const int col  = ntile + l15;
  const int row0 = mtile + ((lane >> 4) << 3);
  float* __restrict__ ep = E + (size_t)row0 * NN + col;
#pragma unroll
  for (int v = 0; v < 8; ++v) {
    ep[(size_t)v * NN] = expf(c[v]);
  }
}

// ---------------------------------------------------------------------------
// Kernel 3: fused softmax-normalize + gate + shift + sharpen + normalize.
// One 256-thread block per batch row; each thread owns 32 strided elements.
// ---------------------------------------------------------------------------
__global__ __launch_bounds__(256) void ntm_epilogue_kernel(
    const float* __restrict__ E, const float* __restrict__ g,
    const float* __restrict__ s, const float* __restrict__ gamma,
    const float* __restrict__ prev, float* __restrict__ out)
{
  __shared__ float gatedS[NN];
  __shared__ float red[256];
  __shared__ float ssS[3];

  const int b = blockIdx.x;
  const int t = threadIdx.x;
  const float* __restrict__ Erow = E + (size_t)b * NN;

  // --- pass 1: load exp row, partial sum
  float e[32];
  float lsum = 0.f;
#pragma unroll
  for (int i = 0; i < 32; ++i) { float v = Erow[i * 256 + t]; e[i] = v; lsum += v; }
  red[t] = lsum;
  for (int s2 = 128; s2 > 0; s2 >>= 1) { __syncthreads(); if (t < s2) red[t] += red[t + s2]; }
  __syncthreads();
  const float sumE = red[0];

  if (t == 0) {  // softmax over the 3 shift logits
    const float s0 = s[b * 3 + 0], s1 = s[b * 3 + 1], s2v = s[b * 3 + 2];
    const float mx = fmaxf(s0, fmaxf(s1, s2v));
    const float e0 = expf(s0 - mx), e1 = expf(s1 - mx), e2 = expf(s2v - mx);
    const float inv = 1.f / (e0 + e1 + e2);
    ssS[0] = e0 * inv; ssS[1] = e1 * inv; ssS[2] = e2 * inv;
  }

  // --- gated = g*softmax + (1-g)*prev, staged in LDS for the shift halo
  const float gb    = g[b];
  const float scale = gb / sumE;
  const float og    = 1.f - gb;
  const float* __restrict__ prow = prev + (size_t)b * NN;
#pragma unroll
  for (int i = 0; i < 32; ++i) {
    const int n = i * 256 + t;
    gatedS[n] = e[i] * scale + og * prow[n];
  }
  __syncthreads();

  // --- 3-tap non-circular shift, then sharpen
  const float ss0 = ssS[0], ss1 = ssS[1], ss2 = ssS[2];
  const float gm  = gamma[b];
  float sh[32];
  float lsum2 = 0.f;
#pragma unroll
  for (int i = 0; i < 32; ++i) {
    const int n = i * 256 + t;
    const float left  = (n > 0)      ? gatedS[n - 1] : 0.f;
    const float right = (n < NN - 1) ? gatedS[n + 1] : 0.f;
    const float sv = left * ss0 + gatedS[n] * ss1 + right * ss2;
    const float p  = powf(sv, gm);
    sh[i] = p; lsum2 += p;
  }
  __syncthreads();
  red[t] = lsum2;
  for (int s2 = 128; s2 > 0; s2 >>= 1) { __syncthreads(); if (t < s2) red[t] += red[t + s2]; }
  __syncthreads();
  const float invS = 1.f / (red[0] + 1e-8f);

  float* __restrict__ orow = out + (size_t)b * NN;
#pragma unroll
  for (int i = 0; i < 32; ++i) orow[i * 256 + t] = sh[i] * invS;
}

// ---------------------------------------------------------------------------
extern "C" void kernel_launch(void* const* d_in, const int* in_sizes, int n_in,
                              void* d_out, int out_size, void* d_ws, size_t ws_size,
                              hipStream_t stream) {
  const float* k      = (const float*)d_in[0];  // [B,W]
  const float* beta   = (const float*)d_in[1];  // [B,1]
  const float* g      = (const float*)d_in[2];  // [B,1]
  const float* s      = (const float*)d_in[3];  // [B,3]
  const float* gamma  = (const float*)d_in[4];  // [B,1]
  const float* prev   = (const float*)d_in[5];  // [B,N]
  const float* memory = (const float*)d_in[6];  // [N,W]
  float* out = (float*)d_out;

  char* ws = (char*)d_ws;
  float* E   = (float*)ws;                                            // B*N f32 (128 MB)
  float* Asc = (float*)(ws + (size_t)BB * NN * sizeof(float));        // B*W
  float* Mc  = (float*)(ws + (size_t)BB * NN * sizeof(float)
                           + (size_t)BB * WW * sizeof(float));        // N*W

  ntm_prep_kernel<<<BB + NN, 64, 0, stream>>>(k, beta, memory, Asc, Mc);
  ntm_gemm_exp_kernel<<<dim3(NN / 128, BB / 16), 256, 0, stream>>>(Asc, Mc, E);
  ntm_epilogue_kernel<<<BB, 256, 0, stream>>>(E, g, s, gamma, prev, out);
}